// VqNet_22256520528420
// MI455X (gfx1250) — compile-verified
//
#include <hip/hip_runtime.h>
#include <hip/hip_bf16.h>
#include <math.h>

// CDNA5 / gfx1250, wave32.
typedef __attribute__((ext_vector_type(2))) float v2f;
typedef __attribute__((ext_vector_type(8))) float v8f;

#define KCLS 16

// ---------------------------------------------------------------------------
// Kernel A: zero the accumulators (qz region = D accumulator, Vq region = S_lo).
// d_out is poisoned once by the harness and not re-poisoned between replays,
// so every launch must re-zero.
// ---------------------------------------------------------------------------
__global__ void vq_zero_kernel(float* __restrict__ out, int n_total) {
  int n4 = n_total >> 2;
  int t = blockIdx.x * blockDim.x + threadIdx.x;
  if (t < n4) {
    ((float4*)out)[t] = make_float4(0.f, 0.f, 0.f, 0.f);
  }
  if (t == 0) {
    for (int k = n4 << 2; k < n_total; ++k) out[k] = 0.f;
  }
}

// ---------------------------------------------------------------------------
// Kernel B: per-j table. lo[j] = log(nsig/16), diff[j] = log(sig + nsig/16) - lo.
// J = 5000 -> 40KB table, stays hot in cache for the scatter pass.
// ---------------------------------------------------------------------------
__global__ void vq_table_kernel(const float* __restrict__ snr,
                                float2* __restrict__ tab, int J) {
  int j = blockIdx.x * blockDim.x + threadIdx.x;
  if (j >= J) return;
  float x    = snr[j];
  float sig  = 1.0f / (1.0f + __expf(-x));
  float nsig = 1.0f / (1.0f + __expf(x));
  float off  = nsig * 0.0625f;          // nsig / K
  float lo   = __logf(off);
  float ld   = __logf(sig + off);
  tab[j] = make_float2(lo, ld - lo);
}

// ---------------------------------------------------------------------------
// Kernel C: scatter. Two native fp32 global atomics per observation:
//   D[ii[n], y[n]] += diff[jj[n]]   (one-hot part)
//   S_lo[ii[n]]    += lo[jj[n]]     (scalar part, folded into Vq later)
// Accumulators live in d_out (34MB -> fully resident in the 192MB L2, so the
// atomic RMW traffic never reaches HBM). unsafeAtomicAdd emits the native
// device-scope global_atomic_add_f32 (no-return form, STOREcnt tracked).
// ---------------------------------------------------------------------------
__global__ void vq_scatter_kernel(const int* __restrict__ ii,
                                  const int* __restrict__ jj,
                                  const int* __restrict__ y,
                                  const float2* __restrict__ tab,
                                  float* __restrict__ d_acc,   // I*16
                                  float* __restrict__ slo_acc, // I
                                  int N) {
  int n = blockIdx.x * blockDim.x + threadIdx.x;
  if (n >= N) return;
  int i = ii[n];
  int j = jj[n];
  int a = y[n];
  float2 t = tab[j];
  unsafeAtomicAdd(&d_acc[(size_t)i * KCLS + a], t.y);
  unsafeAtomicAdd(&slo_acc[i], t.x);
}

// ---------------------------------------------------------------------------
// Kernel D: finalize. One wave handles a tile of 16 rows; lane L owns row
// tile*16 + (L&15) (both half-waves mirror). Per-lane serial max + exp over
// its 16 contiguous values, then V_WMMA_F32_16X16X4_F32 with A=ones performs
// the 16-wide cross-lane reduction: D[m,n] = sum_k B[k,n], so after 4
// accumulating K=4 steps every lane holds the softmax denominator of its own
// row in acc[0]. In-place: qz overwrites D, Vq = S_lo + m + log(sum).
// EXEC stays all-ones (selects only) until after the last WMMA.
// s >= 1 always (the max element contributes exp(0)=1), so a single
// v_rcp_f32 is accurate and avoids the 12-op IEEE divide sequence.
// ---------------------------------------------------------------------------
__global__ void vq_finalize_kernel(float* __restrict__ qz,
                                   float* __restrict__ vq, int I_) {
  const int lane = threadIdx.x & 31;
  const int wave = (int)((blockIdx.x * (unsigned)blockDim.x + threadIdx.x) >> 5);
  const int nrow = lane & 15;
  const int r    = wave * 16 + nrow;
  const bool valid = (r < I_);
  const int rr = valid ? r : 0;

  const float4* __restrict__ p = (const float4*)(qz + (size_t)rr * KCLS);
  float4 q0 = p[0], q1 = p[1], q2 = p[2], q3 = p[3];
  float x[16] = {q0.x, q0.y, q0.z, q0.w, q1.x, q1.y, q1.z, q1.w,
                 q2.x, q2.y, q2.z, q2.w, q3.x, q3.y, q3.z, q3.w};

  float m = x[0];
#pragma unroll
  for (int k = 1; k < 16; ++k) m = fmaxf(m, x[k]);

  float e[16];
#pragma unroll
  for (int k = 0; k < 16; ++k) e[k] = __expf(x[k] - m);

  // B layout for 32-bit 4x16: lane&15 = N (column = our row), VGPR picks K;
  // lanes 0-15 carry K={4c,4c+1}, lanes 16-31 carry K={4c+2,4c+3}.
  const bool hi = (lane >= 16);
  v2f a; a.x = 1.0f; a.y = 1.0f;   // A = ones(16x4)
  v8f acc = {};
#pragma unroll
  for (int cb = 0; cb < 4; ++cb) {
    v2f b;
    b.x = hi ? e[4 * cb + 2] : e[4 * cb + 0];
    b.y = hi ? e[4 * cb + 3] : e[4 * cb + 1];
    acc = __builtin_amdgcn_wmma_f32_16x16x4_f32(
        /*neg_a=*/false, a, /*neg_b=*/false, b,
        /*c_mod=*/(short)0, acc, /*reuse_a=*/false, /*reuse_b=*/false);
  }
  float s   = acc[0];                      // rowsum of exp for row (lane&15)
  float inv = __builtin_amdgcn_rcpf(s);    // v_rcp_f32; s in [1, 16], safe
  float lse = m + __logf(s);

  if (valid && lane < 16) {    // low half writes; high half was a mirror
    float4* o = (float4*)(qz + (size_t)r * KCLS);
    o[0] = make_float4(e[0] * inv,  e[1] * inv,  e[2] * inv,  e[3] * inv);
    o[1] = make_float4(e[4] * inv,  e[5] * inv,  e[6] * inv,  e[7] * inv);
    o[2] = make_float4(e[8] * inv,  e[9] * inv,  e[10] * inv, e[11] * inv);
    o[3] = make_float4(e[12] * inv, e[13] * inv, e[14] * inv, e[15] * inv);
    vq[r] += lse;              // vq currently holds S_lo[r]
  }
}

// ---------------------------------------------------------------------------
extern "C" void kernel_launch(void* const* d_in, const int* in_sizes, int n_in,
                              void* d_out, int out_size, void* d_ws, size_t ws_size,
                              hipStream_t stream) {
  const float* snr = (const float*)d_in[0];
  const int*   ii  = (const int*)d_in[1];
  const int*   jj  = (const int*)d_in[2];
  const int*   y   = (const int*)d_in[3];
  const int J = in_sizes[0];
  const int N = in_sizes[1];
  const int I = out_size / (KCLS + 1);   // out = qz (I*16) ++ Vq (I)

  float*  qz  = (float*)d_out;
  float*  vq  = qz + (size_t)I * KCLS;
  float2* tab = (float2*)d_ws;           // needs 8*J = 40KB scratch

  // A: zero accumulators (entire d_out)
  {
    int n4 = out_size >> 2;
    int blocks = (n4 + 255) / 256;
    if (blocks < 1) blocks = 1;
    vq_zero_kernel<<<blocks, 256, 0, stream>>>((float*)d_out, out_size);
  }
  // B: per-j {lo, diff} table
  {
    int blocks = (J + 255) / 256;
    vq_table_kernel<<<blocks, 256, 0, stream>>>(snr, tab, J);
  }
  // C: scatter with fp32 global atomics
  {
    int blocks = (N + 255) / 256;
    vq_scatter_kernel<<<blocks, 256, 0, stream>>>(ii, jj, y, tab, qz, vq, N);
  }
  // D: WMMA-assisted softmax/LSE finalize, 16 rows per wave
  {
    long long tiles   = (I + 15) / 16;
    long long threads = tiles * 32;
    int blocks = (int)((threads + 255) / 256);
    if (blocks < 1) blocks = 1;
    vq_finalize_kernel<<<blocks, 256, 0, stream>>>(qz, vq, I);
  }
}